// SurrogateModel_54099408060634
// MI455X (gfx1250) — compile-verified
//
#include <hip/hip_runtime.h>

// ---- WMMA vector types (wave32, gfx1250) ----
typedef float v2f __attribute__((ext_vector_type(2)));
typedef float v8f __attribute__((ext_vector_type(8)));

#define IN_DIM 128
#define HID    64
#define OUT_D  16

// ============================================================
// Degree / normalization
// ============================================================
__global__ void k_init_deg(float* __restrict__ deg, int n) {
    int i = blockIdx.x * blockDim.x + threadIdx.x;
    if (i < n) deg[i] = 1.0f;               // self-loop contributes 1
}

__global__ void k_deg_accum(const int* __restrict__ dst, float* __restrict__ deg, int E) {
    int e = blockIdx.x * blockDim.x + threadIdx.x;
    if (e < E) atomicAdd(&deg[dst[e]], 1.0f);
}

__global__ void k_dinv(float* __restrict__ deg, int n) {
    int i = blockIdx.x * blockDim.x + threadIdx.x;
    if (i < n) deg[i] = rsqrtf(deg[i]);     // deg >= 1 always (self loop)
}

// ============================================================
// GEMM1: h1[n,64] = x[n,128] @ W1[128,64]   (fp32 WMMA 16x16x4)
// block = 128 threads = 4 waves; one 16-row M-tile per block,
// wave w computes N-columns [16w, 16w+16).
// ============================================================
#define XS_STRIDE 132   // 16 rows of x tile; 4*m+k banks -> conflict-free
#define W1_STRIDE 72    // 2-row group shift = 16 banks -> disjoint half-waves

__global__ void __launch_bounds__(128)
k_gemm1(const float* __restrict__ x, const float* __restrict__ W1,
        float* __restrict__ h1, int n) {
    __shared__ float Ws[IN_DIM * W1_STRIDE];   // ~36.9 KB
    __shared__ float xs[16 * XS_STRIDE];       // ~8.4 KB
    const int tid  = threadIdx.x;
    const int row0 = blockIdx.x * 16;

    for (int i = tid; i < IN_DIM * HID; i += 128) {
        int k = i >> 6, c = i & 63;
        Ws[k * W1_STRIDE + c] = W1[i];
    }
    for (int i = tid; i < 16 * IN_DIM; i += 128) {
        int r = i >> 7, c = i & 127;
        int gr = row0 + r;
        xs[r * XS_STRIDE + c] = (gr < n) ? x[(size_t)gr * IN_DIM + c] : 0.0f;
    }
    __syncthreads();

    const int lane = tid & 31;
    const int wave = tid >> 5;             // N-tile index (0..3)
    const int m    = lane & 15;            // A row
    const int kb   = (lane >> 4) << 1;     // K sub-offset: 0 or 2
    const int ncol = (wave << 4) + (lane & 15);

    v8f acc = {};
    #pragma unroll
    for (int k = 0; k < IN_DIM; k += 4) {
        v2f a, b;
        a.x = xs[m * XS_STRIDE + k + kb];
        a.y = xs[m * XS_STRIDE + k + kb + 1];
        b.x = Ws[(k + kb)     * W1_STRIDE + ncol];
        b.y = Ws[(k + kb + 1) * W1_STRIDE + ncol];
        acc = __builtin_amdgcn_wmma_f32_16x16x4_f32(false, a, false, b,
                                                    (short)0, acc, false, false);
    }
    const int rhalf = (lane >> 4) << 3;    // 0 or 8
    #pragma unroll
    for (int g = 0; g < 8; ++g) {
        int r = row0 + rhalf + g;
        if (r < n) h1[(size_t)r * HID + ncol] = acc[g];
    }
}

// ============================================================
// out1[i][j] = b1[j] + h1[i][j]*dinv[i]^2   (bias + self-loop term;
// also serves as accumulator init for the edge scatter)
// ============================================================
__global__ void k_init_out1(const float* __restrict__ h1, const float* __restrict__ dinv,
                            const float* __restrict__ b1, float* __restrict__ out1, int n) {
    int idx = blockIdx.x * blockDim.x + threadIdx.x;
    if (idx >= n * HID) return;
    int i = idx >> 6, j = idx & 63;
    float di = dinv[i];
    out1[idx] = b1[j] + h1[idx] * di * di;
}

// 4 threads per edge, 16 columns each: float4 gathers from L2-resident h1,
// fp32 atomics into L2-resident out1.
__global__ void k_scatter1(const int* __restrict__ src, const int* __restrict__ dst,
                           const float* __restrict__ h1, const float* __restrict__ dinv,
                           float* __restrict__ out1, int E) {
    int t = blockIdx.x * blockDim.x + threadIdx.x;
    if (t >= E * 4) return;
    int e = t >> 2, q = t & 3;
    int s = src[e], d = dst[e];
    float w = dinv[s] * dinv[d];
    const float4* hp = (const float4*)(h1 + (size_t)s * HID + q * 16);
    float*        op = out1 + (size_t)d * HID + q * 16;
    #pragma unroll
    for (int v = 0; v < 4; ++v) {
        float4 mv = hp[v];
        atomicAdd(op + v * 4 + 0, mv.x * w);
        atomicAdd(op + v * 4 + 1, mv.y * w);
        atomicAdd(op + v * 4 + 2, mv.z * w);
        atomicAdd(op + v * 4 + 3, mv.w * w);
    }
}

// ============================================================
// GEMM2: h2[n,16] = relu(out1)[n,64] @ W2[64,16]  (ReLU fused into A staging)
// block = 128 threads = 4 waves; 64 rows per block, one M-tile per wave.
// ============================================================
#define AS_STRIDE 68    // 4*mrow + k + kb banks -> conflict-free
#define W2_STRIDE 24    // 2-row group shift = 48 banks -> disjoint half-waves

__global__ void __launch_bounds__(128)
k_gemm2(const float* __restrict__ out1, const float* __restrict__ W2,
        float* __restrict__ h2, int n) {
    __shared__ float Ws[HID * W2_STRIDE];   // 6 KB
    __shared__ float as[64 * AS_STRIDE];    // 17 KB
    const int tid  = threadIdx.x;
    const int row0 = blockIdx.x * 64;

    for (int i = tid; i < HID * OUT_D; i += 128) {
        int k = i >> 4, c = i & 15;
        Ws[k * W2_STRIDE + c] = W2[i];
    }
    for (int i = tid; i < 64 * HID; i += 128) {
        int r = i >> 6, c = i & 63;
        int gr = row0 + r;
        as[r * AS_STRIDE + c] = (gr < n) ? fmaxf(out1[(size_t)gr * HID + c], 0.0f) : 0.0f;
    }
    __syncthreads();

    const int lane = tid & 31;
    const int wave = tid >> 5;
    const int m    = lane & 15;
    const int kb   = (lane >> 4) << 1;
    const int ncol = lane & 15;
    const int mrow = wave * 16 + m;

    v8f acc = {};
    #pragma unroll
    for (int k = 0; k < HID; k += 4) {
        v2f a, b;
        a.x = as[mrow * AS_STRIDE + k + kb];
        a.y = as[mrow * AS_STRIDE + k + kb + 1];
        b.x = Ws[(k + kb)     * W2_STRIDE + ncol];
        b.y = Ws[(k + kb + 1) * W2_STRIDE + ncol];
        acc = __builtin_amdgcn_wmma_f32_16x16x4_f32(false, a, false, b,
                                                    (short)0, acc, false, false);
    }
    const int rhalf = (lane >> 4) << 3;
    #pragma unroll
    for (int g = 0; g < 8; ++g) {
        int r = row0 + wave * 16 + rhalf + g;
        if (r < n) h2[(size_t)r * OUT_D + ncol] = acc[g];
    }
}

// ============================================================
// Layer-2 bias + self-loop init, then edge scatter (16 cols/edge)
// ============================================================
__global__ void k_init_out2(const float* __restrict__ h2, const float* __restrict__ dinv,
                            const float* __restrict__ b2, float* __restrict__ out, int n) {
    int idx = blockIdx.x * blockDim.x + threadIdx.x;
    if (idx >= n * OUT_D) return;
    int i = idx >> 4, j = idx & 15;
    float di = dinv[i];
    out[idx] = b2[j] + h2[idx] * di * di;
}

__global__ void k_scatter2(const int* __restrict__ src, const int* __restrict__ dst,
                           const float* __restrict__ h2, const float* __restrict__ dinv,
                           float* __restrict__ out, int E) {
    int e = blockIdx.x * blockDim.x + threadIdx.x;
    if (e >= E) return;
    int s = src[e], d = dst[e];
    float w = dinv[s] * dinv[d];
    const float4* hp = (const float4*)(h2 + (size_t)s * OUT_D);
    float*        op = out + (size_t)d * OUT_D;
    #pragma unroll
    for (int v = 0; v < 4; ++v) {
        float4 mv = hp[v];
        atomicAdd(op + v * 4 + 0, mv.x * w);
        atomicAdd(op + v * 4 + 1, mv.y * w);
        atomicAdd(op + v * 4 + 2, mv.z * w);
        atomicAdd(op + v * 4 + 3, mv.w * w);
    }
}

// ============================================================
extern "C" void kernel_launch(void* const* d_in, const int* in_sizes, int n_in,
                              void* d_out, int out_size, void* d_ws, size_t ws_size,
                              hipStream_t stream) {
    const float* x   = (const float*)d_in[0];
    const int*   ei  = (const int*)d_in[1];
    const float* W1  = (const float*)d_in[2];
    const float* b1  = (const float*)d_in[3];
    const float* W2  = (const float*)d_in[4];
    const float* b2  = (const float*)d_in[5];
    float*       out = (float*)d_out;

    const int n = in_sizes[0] / IN_DIM;        // 100000
    const int E = in_sizes[1] / 2;             // 1600000
    const int* src = ei;
    const int* dst = ei + E;

    // Workspace layout (all fp32): deg/dinv | h1 (reused as h2) | out1
    float* deg  = (float*)d_ws;
    float* h1   = deg + ((n + 63) / 64) * 64;  // keep 256B alignment
    float* out1 = h1 + (size_t)n * HID;
    float* h2   = h1;                          // h1 dead after scatter1

    const int B = 256;

    // degree -> dinv
    k_init_deg <<<(n + B - 1) / B, B, 0, stream>>>(deg, n);
    k_deg_accum<<<(E + B - 1) / B, B, 0, stream>>>(dst, deg, E);
    k_dinv     <<<(n + B - 1) / B, B, 0, stream>>>(deg, n);

    // layer 1
    k_gemm1    <<<(n + 15) / 16, 128, 0, stream>>>(x, W1, h1, n);
    k_init_out1<<<((n * HID) + B - 1) / B, B, 0, stream>>>(h1, deg, b1, out1, n);
    k_scatter1 <<<((E * 4) + B - 1) / B, B, 0, stream>>>(src, dst, h1, deg, out1, E);

    // layer 2 (ReLU fused into gemm2 A staging)
    k_gemm2    <<<(n + 63) / 64, 128, 0, stream>>>(out1, W2, h2, n);
    k_init_out2<<<((n * OUT_D) + B - 1) / B, B, 0, stream>>>(h2, deg, b2, out, n);
    k_scatter2 <<<(E + B - 1) / B, B, 0, stream>>>(src, dst, h2, deg, out, E);
}